// PrototypicalNetwork_22505628631296
// MI455X (gfx1250) — compile-verified
//
#include <hip/hip_runtime.h>
#include <hip/hip_bf16.h>

typedef __attribute__((ext_vector_type(2))) float v2f;
typedef __attribute__((ext_vector_type(8))) float v8f;

#define NTHREADS 256
#define NWAVES   8
#define NUM_WAYS 20

// -------- workspace layout (float offsets) --------
#define RW1_OFF 0                         // 4*9*32*2   = 2304 floats (Cin=3)
#define RW2_OFF 2304                      // 4*192*32*2 = 49152 floats
#define RW3_OFF (RW2_OFF + 49152)
#define RW4_OFF (RW3_OFF + 49152)
#define E_OFF   (RW4_OFF + 49152)         // 1124*20 embeddings
#define D_OFF   (E_OFF + 1124*NUM_WAYS)   // 20 diag prototypes

// ---------------------------------------------------------------------------
// Repack conv weights (Cout=64, Cin, 3, 3) into WMMA A-fragment order.
// K per input channel padded 9 -> 12 taps (pads get weight 0) so each
// 16x16x4 f32 WMMA k-step stays inside one channel.
// A(16x4) layout: lane l -> M = l&15 ; K = vgpr + 2*(l>>4).
// ---------------------------------------------------------------------------
__global__ void repack_kernel(const float* __restrict__ w, int Cin,
                              v2f* __restrict__ rw) {
  int idx = blockIdx.x * blockDim.x + threadIdx.x;
  int K3 = Cin * 3;
  int total = 4 * K3 * 32;
  if (idx >= total) return;
  int lane  = idx & 31;
  int kstep = (idx >> 5) % K3;
  int ct    = (idx >> 5) / K3;
  int ci    = kstep / 3;
  int s     = kstep - 3 * ci;
  int m     = lane & 15;
  int h     = lane >> 4;
  int cout  = ct * 16 + m;
  int t0    = s * 4 + 2 * h;
  float a0 = (t0     < 9) ? w[(cout * Cin + ci) * 9 + t0]     : 0.0f;
  float a1 = (t0 + 1 < 9) ? w[(cout * Cin + ci) * 9 + t0 + 1] : 0.0f;
  v2f r; r.x = a0; r.y = a1;
  rw[idx] = r;
}

__device__ __forceinline__ void zero_lds(float* __restrict__ p, int n, int tid) {
  for (int i = tid; i < n; i += NTHREADS) p[i] = 0.f;
}

// One conv layer. Input lin is HALO-PADDED in LDS: each channel stored as
// (H+2)x(W+2) with zero border, so tap reads are branch-free (no exec
// masking around ds_load in the WMMA k-loop). Output lout is unpadded
// 64 x H x W (bias + ReLU applied). One wave owns a 64(cout) x 16(pixel)
// tile: 4 f32 accumulators, K-loop of Cin*3 WMMA k-steps x 4 cout tiles.
template<int CIN, int H, int W>
__device__ __forceinline__ void conv_layer(const float* __restrict__ lin,
                                           float* __restrict__ lout,
                                           const v2f* __restrict__ rw,
                                           const float* __restrict__ sbias,
                                           int lane, int wave) {
  constexpr int HW = H * W;
  constexpr int WP = W + 2;
  constexpr int CH = (H + 2) * (W + 2);   // padded channel stride
  constexpr int K3 = CIN * 3;
  constexpr int NT = (HW + 15) / 16;
  const int p = lane & 15;
  const int h = lane >> 4;
  for (int tile = wave; tile < NT; tile += NWAVES) {
    int  pix    = tile * 16 + p;
    bool pvalid = pix < HW;
    int  pc     = pvalid ? pix : (HW - 1);
    int  y      = pc / W;
    int  x      = pc - y * W;
    // Per-lane tap offsets into the padded channel; invariant over ci.
    // B(4x16): K = vgpr + 2*(lane>>4); taps t>=9 are zero-weight pads
    // (clamped to tap 0, killed by zero A).
    int o[6];
#pragma unroll
    for (int s = 0; s < 3; ++s) {
      int t0  = s * 4 + 2 * h;
      int tt0 = (t0     < 9) ? t0     : 0;
      int tt1 = (t0 + 1 < 9) ? t0 + 1 : 0;
      o[2 * s    ] = (y + tt0 / 3) * WP + x + (tt0 - 3 * (tt0 / 3));
      o[2 * s + 1] = (y + tt1 / 3) * WP + x + (tt1 - 3 * (tt1 / 3));
    }
    v8f a0 = {0.f,0.f,0.f,0.f,0.f,0.f,0.f,0.f};
    v8f a1 = {0.f,0.f,0.f,0.f,0.f,0.f,0.f,0.f};
    v8f a2 = {0.f,0.f,0.f,0.f,0.f,0.f,0.f,0.f};
    v8f a3 = {0.f,0.f,0.f,0.f,0.f,0.f,0.f,0.f};
#pragma unroll 1
    for (int ci = 0; ci < CIN; ++ci) {
      const float* ch = lin + ci * CH;
#pragma unroll
      for (int s = 0; s < 3; ++s) {
        v2f B;
        B.x = ch[o[2 * s]];
        B.y = ch[o[2 * s + 1]];
        int kb = (ci * 3 + s) * 32 + lane;
        v2f A0 = rw[kb];
        v2f A1 = rw[kb + 1 * K3 * 32];
        v2f A2 = rw[kb + 2 * K3 * 32];
        v2f A3 = rw[kb + 3 * K3 * 32];
        a0 = __builtin_amdgcn_wmma_f32_16x16x4_f32(false, A0, false, B, (short)0, a0, false, false);
        a1 = __builtin_amdgcn_wmma_f32_16x16x4_f32(false, A1, false, B, (short)0, a1, false, false);
        a2 = __builtin_amdgcn_wmma_f32_16x16x4_f32(false, A2, false, B, (short)0, a2, false, false);
        a3 = __builtin_amdgcn_wmma_f32_16x16x4_f32(false, A3, false, B, (short)0, a3, false, false);
      }
    }
    if (pvalid) {
#pragma unroll
      for (int v = 0; v < 8; ++v) {
        int co = v + 8 * h;                // C/D: M = vgpr + 8*(lane>>4)
        float r0 = a0[v] + sbias[co];
        float r1 = a1[v] + sbias[co + 16];
        float r2 = a2[v] + sbias[co + 32];
        float r3 = a3[v] + sbias[co + 48];
        lout[(co     ) * HW + pix] = r0 > 0.f ? r0 : 0.f;
        lout[(co + 16) * HW + pix] = r1 > 0.f ? r1 : 0.f;
        lout[(co + 32) * HW + pix] = r2 > 0.f ? r2 : 0.f;
        lout[(co + 48) * HW + pix] = r3 > 0.f ? r3 : 0.f;
      }
    }
  }
}

// 2x2 floor max-pool, 64 channels. Input: unpadded HxW conv output.
// Output: written into the HALO-PADDED interior of the next layer's input
// buffer ((Hp+2)x(Wp+2) per channel; border pre-zeroed by caller).
template<int H, int W>
__device__ __forceinline__ void pool2(const float* __restrict__ in,
                                      float* __restrict__ out, int tid) {
  constexpr int Hp = H / 2, Wp = W / 2;
  constexpr int CHP = (Hp + 2) * (Wp + 2);
  constexpr int tot = 64 * Hp * Wp;
  for (int i = tid; i < tot; i += NTHREADS) {
    int co = i / (Hp * Wp);
    int r  = i - co * (Hp * Wp);
    int py = r / Wp;
    int px = r - py * Wp;
    const float* b = in + co * (H * W) + (2 * py) * W + 2 * px;
    out[co * CHP + (py + 1) * (Wp + 2) + (px + 1)] =
        fmaxf(fmaxf(b[0], b[1]), fmaxf(b[W], b[W + 1]));
  }
}

// Whole Conv4 backbone + linear head, one image per block, LDS-resident:
// bufA = 64*28*28 conv scratch (200KB), bufB = max padded input 64*16*16
// (64KB), + 64 bias floats => ~260KB of the 320KB/WGP.
__global__ __launch_bounds__(NTHREADS)
void conv4_fwd(const float* __restrict__ x,
               const v2f* __restrict__ rw1, const v2f* __restrict__ rw2,
               const v2f* __restrict__ rw3, const v2f* __restrict__ rw4,
               const float* __restrict__ b1, const float* __restrict__ b2,
               const float* __restrict__ b3, const float* __restrict__ b4,
               const float* __restrict__ wf, const float* __restrict__ bf,
               float* __restrict__ e_out) {
  extern __shared__ float smem[];
  float* bufA = smem;                          // 64*28*28 = 50176 floats
  float* bufB = smem + 64 * 28 * 28;           // 64*16*16 = 16384 floats
  float* sb   = bufB + 64 * 16 * 16;           // 64 floats (bias stage)
  const int n    = blockIdx.x;
  const int tid  = threadIdx.x;
  const int lane = tid & 31;
  const int wave = tid >> 5;

  // Stage image into padded 3 x 30x30 layout (zero halo).
  const float* img = x + (size_t)n * (3 * 28 * 28);
  zero_lds(bufB, 3 * 30 * 30, tid);
  if (tid < 64) sb[tid] = b1[tid];
  __syncthreads();
  for (int i = tid; i < 3 * 28 * 28; i += NTHREADS) {
    int ci = i / 784; int r = i - ci * 784; int y = r / 28; int x0 = r - y * 28;
    bufB[ci * 900 + (y + 1) * 30 + (x0 + 1)] = img[i];
  }
  __syncthreads();
  conv_layer<3, 28, 28>(bufB, bufA, rw1, sb, lane, wave);
  __syncthreads();
  zero_lds(bufB, 64 * 16 * 16, tid);           // padded 14x14 -> 16x16
  if (tid < 64) sb[tid] = b2[tid];
  __syncthreads();
  pool2<28, 28>(bufA, bufB, tid);
  __syncthreads();
  conv_layer<64, 14, 14>(bufB, bufA, rw2, sb, lane, wave);
  __syncthreads();
  zero_lds(bufB, 64 * 9 * 9, tid);             // padded 7x7 -> 9x9
  if (tid < 64) sb[tid] = b3[tid];
  __syncthreads();
  pool2<14, 14>(bufA, bufB, tid);
  __syncthreads();
  conv_layer<64, 7, 7>(bufB, bufA, rw3, sb, lane, wave);
  __syncthreads();
  zero_lds(bufB, 64 * 5 * 5, tid);             // padded 3x3 -> 5x5
  if (tid < 64) sb[tid] = b4[tid];
  __syncthreads();
  pool2<7, 7>(bufA, bufB, tid);
  __syncthreads();
  conv_layer<64, 3, 3>(bufB, bufA, rw4, sb, lane, wave);
  __syncthreads();
  pool2<3, 3>(bufA, bufB, tid);                // 1x1 at padded slot c*9+4
  __syncthreads();
  if (tid < NUM_WAYS) {                        // linear head [64]->[20]
    float acc = bf[tid];
#pragma unroll 1
    for (int c = 0; c < 64; ++c) acc += bufB[c * 9 + 4] * wf[c * NUM_WAYS + tid];
    e_out[(size_t)n * NUM_WAYS + tid] = acc;
  }
}

// d[j] = mean over support i with y==j of e_supp[i][j]  (diagonal of protos)
__global__ void proto_diag_kernel(const float* __restrict__ e_supp,
                                  const int* __restrict__ y_supp,
                                  int Ns, float* __restrict__ d) {
  int j = threadIdx.x;
  if (j < NUM_WAYS) {
    float s = 0.f, c = 0.f;
    for (int i = 0; i < Ns; ++i) {
      if (y_supp[i] == j) { s += e_supp[i * NUM_WAYS + j]; c += 1.f; }
    }
    d[j] = s / c;
  }
}

// preds[i][j] = -sqrt(sqrt((d[j] - e_q[i][j])^2))
__global__ void preds_kernel(const float* __restrict__ e_q,
                             const float* __restrict__ d,
                             float* __restrict__ out, int total) {
  int idx = blockIdx.x * blockDim.x + threadIdx.x;
  if (idx < total) {
    int j = idx % NUM_WAYS;
    float diff = d[j] - e_q[idx];
    out[idx] = -sqrtf(sqrtf(diff * diff));
  }
}

// loss = -mean_i log_softmax(preds[i])[y_q[i]]
__global__ void loss_kernel(const float* __restrict__ preds,
                            const int* __restrict__ yq,
                            int Nq, float* __restrict__ out_loss) {
  __shared__ float red[NTHREADS];
  float acc = 0.f;
  for (int i = threadIdx.x; i < Nq; i += NTHREADS) {
    const float* row = preds + i * NUM_WAYS;
    float m = row[0];
    for (int j = 1; j < NUM_WAYS; ++j) m = fmaxf(m, row[j]);
    float s = 0.f;
    for (int j = 0; j < NUM_WAYS; ++j) s += expf(row[j] - m);
    float lse = m + logf(s);
    acc += row[yq[i]] - lse;
  }
  red[threadIdx.x] = acc;
  __syncthreads();
  for (int st = NTHREADS / 2; st > 0; st >>= 1) {
    if (threadIdx.x < st) red[threadIdx.x] += red[threadIdx.x + st];
    __syncthreads();
  }
  if (threadIdx.x == 0) out_loss[0] = -red[0] / (float)Nq;
}

extern "C" void kernel_launch(void* const* d_in, const int* in_sizes, int n_in,
                              void* d_out, int out_size, void* d_ws, size_t ws_size,
                              hipStream_t stream) {
  (void)in_sizes; (void)n_in; (void)out_size; (void)ws_size;
  const float* x_supp  = (const float*)d_in[0];
  const int*   y_supp  = (const int*)  d_in[1];
  const float* x_query = (const float*)d_in[2];
  const int*   y_query = (const int*)  d_in[3];
  const float* w1 = (const float*)d_in[4];
  const float* b1 = (const float*)d_in[5];
  const float* w2 = (const float*)d_in[6];
  const float* b2 = (const float*)d_in[7];
  const float* w3 = (const float*)d_in[8];
  const float* b3 = (const float*)d_in[9];
  const float* w4 = (const float*)d_in[10];
  const float* b4 = (const float*)d_in[11];
  const float* wf = (const float*)d_in[12];
  const float* bf = (const float*)d_in[13];

  float* ws  = (float*)d_ws;
  v2f*  rw1  = (v2f*)(ws + RW1_OFF);
  v2f*  rw2  = (v2f*)(ws + RW2_OFF);
  v2f*  rw3  = (v2f*)(ws + RW3_OFF);
  v2f*  rw4  = (v2f*)(ws + RW4_OFF);
  float* e   = ws + E_OFF;
  float* dd  = ws + D_OFF;
  float* preds = (float*)d_out;

  // 1) repack weights into WMMA A-fragment order
  repack_kernel<<<dim3((4 * 9 * 32 + 255) / 256),   dim3(256), 0, stream>>>(w1, 3,  rw1);
  repack_kernel<<<dim3((4 * 192 * 32 + 255) / 256), dim3(256), 0, stream>>>(w2, 64, rw2);
  repack_kernel<<<dim3((4 * 192 * 32 + 255) / 256), dim3(256), 0, stream>>>(w3, 64, rw3);
  repack_kernel<<<dim3((4 * 192 * 32 + 255) / 256), dim3(256), 0, stream>>>(w4, 64, rw4);

  // 2) backbone + head (one block per image, LDS-resident)
  size_t smem_bytes = (size_t)(64 * 28 * 28 + 64 * 16 * 16 + 64) * sizeof(float);
  conv4_fwd<<<dim3(100),  dim3(NTHREADS), smem_bytes, stream>>>(
      x_supp,  rw1, rw2, rw3, rw4, b1, b2, b3, b4, wf, bf, e);
  conv4_fwd<<<dim3(1024), dim3(NTHREADS), smem_bytes, stream>>>(
      x_query, rw1, rw2, rw3, rw4, b1, b2, b3, b4, wf, bf, e + 100 * NUM_WAYS);

  // 3) diagonal prototypes, 4) preds, 5) loss
  proto_diag_kernel<<<dim3(1), dim3(32), 0, stream>>>(e, y_supp, 100, dd);
  preds_kernel<<<dim3((1024 * NUM_WAYS + 255) / 256), dim3(256), 0, stream>>>(
      e + 100 * NUM_WAYS, dd, preds, 1024 * NUM_WAYS);
  loss_kernel<<<dim3(1), dim3(NTHREADS), 0, stream>>>(
      preds, y_query, 1024, preds + 1024 * NUM_WAYS);
}